// X_debaising_5085241279109
// MI455X (gfx1250) — compile-verified
//
#include <hip/hip_runtime.h>
#include <stdint.h>

typedef float v4f __attribute__((ext_vector_type(4)));

#define ROWS_PER_CHUNK 4
#define WAVES_PER_BLOCK 8
#define BLOCK_THREADS (WAVES_PER_BLOCK * 32)
#define ROW_BYTES 512 /* 128 fp32 */

// CDNA5 async copy: global memory -> LDS, tracked by ASYNCcnt.
// %0 = LDS byte offset (VGPR), %1 = 64-bit global address (VGPR pair).
__device__ __forceinline__ void async_load_row_b128(uint32_t lds_off, const float* gaddr) {
    asm volatile("global_load_async_to_lds_b128 %0, %1, off"
                 :
                 : "v"(lds_off), "v"((uint64_t)(uintptr_t)gaddr)
                 : "memory");
}

__global__ __launch_bounds__(BLOCK_THREADS)
void colscale_async_kernel(const float* __restrict__ feat,
                           const float* __restrict__ s,
                           float* __restrict__ out,
                           int n_rows, int n_chunks) {
    // Per-wave double buffer: 2 x (4 rows x 512B) = 4KB/wave, 32KB/block.
    __shared__ float smem[WAVES_PER_BLOCK][2][ROWS_PER_CHUNK * 128];

    const int lane   = threadIdx.x & 31;
    const int wib    = threadIdx.x >> 5;                       // wave index in block
    const int waveId = (blockIdx.x * BLOCK_THREADS + threadIdx.x) >> 5;
    const int nWaves = (gridDim.x * BLOCK_THREADS) >> 5;
    const int col    = lane << 2;                              // lane L -> cols 4L..4L+3

    // Column scales live in 4 VGPRs for the entire kernel.
    const v4f s4 = *(const v4f*)(s + col);

    // LDS byte offset = low 32 bits of the generic shared-memory address.
    uint32_t ldsBase[2];
    ldsBase[0] = (uint32_t)(uintptr_t)&smem[wib][0][0] + (uint32_t)(lane * 16);
    ldsBase[1] = (uint32_t)(uintptr_t)&smem[wib][1][0] + (uint32_t)(lane * 16);

    int chunk = waveId;
    if (chunk >= n_chunks) return;   // wave-uniform exit

    // Prologue: stage first chunk into buffer 0.
    {
        const int base = chunk * ROWS_PER_CHUNK;
        #pragma unroll
        for (int r = 0; r < ROWS_PER_CHUNK; ++r) {
            int row = base + r;                   // clamp tail rows: duplicate
            if (row > n_rows - 1) row = n_rows - 1; // loads are harmless, avoids OOB
            async_load_row_b128(ldsBase[0] + r * ROW_BYTES,
                                feat + (size_t)row * 128 + col);
        }
    }

    int buf = 0;
    for (;;) {
        const int  next     = chunk + nWaves;
        const bool has_next = next < n_chunks;

        if (has_next) {
            // Issue next chunk's async loads into the other buffer first
            // (overlaps DMA with this chunk's compute/stores).
            const int base = next * ROWS_PER_CHUNK;
            #pragma unroll
            for (int r = 0; r < ROWS_PER_CHUNK; ++r) {
                int row = base + r;
                if (row > n_rows - 1) row = n_rows - 1;
                async_load_row_b128(ldsBase[buf ^ 1] + r * ROW_BYTES,
                                    feat + (size_t)row * 128 + col);
            }
            // Async loads complete in order: <=4 outstanding means the
            // 4 loads of the *current* chunk have landed in LDS.
            asm volatile("s_wait_asynccnt 4" ::: "memory");
        } else {
            asm volatile("s_wait_asynccnt 0" ::: "memory");
        }

        // Consume current chunk from LDS, scale, stream out (non-temporal:
        // 1GB working set >> 192MB L2, don't write-allocate output lines).
        const int    base = chunk * ROWS_PER_CHUNK;
        const float* lsrc = &smem[wib][buf][0];
        #pragma unroll
        for (int r = 0; r < ROWS_PER_CHUNK; ++r) {
            const int row = base + r;
            if (row < n_rows) {
                v4f f = *(const v4f*)(lsrc + r * 128 + col);
                v4f o = f * s4;
                __builtin_nontemporal_store(o, (v4f*)(out + (size_t)row * 128 + col));
            }
        }

        if (!has_next) break;
        chunk = next;
        buf ^= 1;
    }
}

extern "C" void kernel_launch(void* const* d_in, const int* in_sizes, int n_in,
                              void* d_out, int out_size, void* d_ws, size_t ws_size,
                              hipStream_t stream) {
    const float* feat = (const float*)d_in[0];
    const float* s    = (const float*)d_in[1];
    float*       out  = (float*)d_out;

    const int n_rows   = in_sizes[0] / 128;
    const int n_chunks = (n_rows + ROWS_PER_CHUNK - 1) / ROWS_PER_CHUNK;

    int blocks = (n_chunks + WAVES_PER_BLOCK - 1) / WAVES_PER_BLOCK;
    if (blocks > 4096) blocks = 4096;   // ~32K waves: enough MLP for 23.3 TB/s
    if (blocks < 1)    blocks = 1;

    colscale_async_kernel<<<blocks, BLOCK_THREADS, 0, stream>>>(
        feat, s, out, n_rows, n_chunks);
}